// Proposals_72430328480061
// MI455X (gfx1250) — compile-verified
//
#include <hip/hip_runtime.h>
#include <stdint.h>
#include <stddef.h>

// ---------------------------------------------------------------------------
// RPN proposal pipeline for MI455X (gfx1250, wave32).
// No matmul exists in this op (sorting + min/max IoU); CDNA5-specific paths:
//   - global_load_async_to_lds_b64/b128 + s_wait_asynccnt (ASYNCcnt) staging
//   - LDS-resident bitonic sort (ds ops), LDS bitmask NMS reduce
// NMS is two-phase: fully parallel 32x32-tile suppression-mask build across
// the whole chip, then a single-wave pipelined bitmask reduce.
// ---------------------------------------------------------------------------

namespace {
constexpr int   kLevels = 5;
constexpr int   kA      = 3;
constexpr int   kHh[kLevels]    = {200, 100, 50, 25, 13};
constexpr int   kWw[kLevels]    = {336, 168, 84, 42, 21};
constexpr float kStride[kLevels]= {4.f, 8.f, 16.f, 32.f, 64.f};
constexpr int   kV[kLevels]     = {201600, 50400, 12600, 3150, 819};   // 3*h*w
constexpr int   kP[kLevels]     = {262144, 65536, 16384, 4096, 1024};  // pow2 pad
constexpr int   kLOff[kLevels]  = {0, 262144, 327680, 344064, 348160};
constexpr int   kKeysTotal      = 349184;
constexpr int   kPre    = 2000;
constexpr int   kNC     = 10000;      // 5 * 2000
constexpr int   kNmsPad = 16384;
constexpr int   kMax    = 1000;
constexpr int   kKeepW  = 313;        // ceil(10000/32)
constexpr int   kSlots  = 10;         // ceil(313/32) words per lane
constexpr float kIouThr = 0.7f;
constexpr int   kTile   = 1024;       // LDS bitonic tile
} // namespace

// ---------------- key helpers: monotone float <-> u32 ----------------------
__device__ __forceinline__ unsigned f2key(float f) {
  unsigned u = __float_as_uint(f);
  return (u & 0x80000000u) ? ~u : (u | 0x80000000u);
}
__device__ __forceinline__ float key2f(unsigned s) {
  unsigned u = (s & 0x80000000u) ? (s & 0x7FFFFFFFu) : ~s;
  return __uint_as_float(u);
}

// ---------------- CDNA5 async copies to LDS (ASYNCcnt path) -----------------
// Low 32 bits of a generic pointer into the LDS aperture are the LDS byte
// address (CDNA5 flat->LDS mapping truncates to addr[31:0]).
__device__ __forceinline__ void async_ldst_b64(void* lds_dst, const void* gsrc) {
  unsigned lds_addr = (unsigned)(uintptr_t)lds_dst;
  asm volatile("global_load_async_to_lds_b64 %0, %1, off"
               :: "v"(lds_addr), "v"(gsrc) : "memory");
}
__device__ __forceinline__ void async_ldst_b128(void* lds_dst, const void* gsrc) {
  unsigned lds_addr = (unsigned)(uintptr_t)lds_dst;
  asm volatile("global_load_async_to_lds_b128 %0, %1, off"
               :: "v"(lds_addr), "v"(gsrc) : "memory");
}
__device__ __forceinline__ void wait_async0() {
  asm volatile("s_wait_asynccnt 0" ::: "memory");
}

// ---------------- 1) per-level sort keys ------------------------------------
// flattened score index (h,w,A): t = y*(W*A) + x*A + a
__global__ void make_keys_kernel(const float* __restrict__ cls,
                                 unsigned long long* __restrict__ keys,
                                 int V, int P, int H, int W) {
  int t = blockIdx.x * blockDim.x + threadIdx.x;
  if (t >= P) return;
  unsigned long long key = 0ull;                 // padding sorts strictly last
  if (t < V) {
    int a = t % kA;
    int r = t / kA;
    int x = r % W;
    int y = r / W;
    float logit = cls[a * (H * W) + y * W + x];
    key = ((unsigned long long)f2key(logit) << 32) |
          (unsigned)(0xFFFFFFFFu - (unsigned)t);  // ties -> lower idx first
  }
  keys[t] = key;
}

// ---------------- 2) bitonic sort (descending by u64 key) -------------------
__global__ void bitonic_global_kernel(unsigned long long* __restrict__ d,
                                      int n, int j, int k) {
  int i = blockIdx.x * blockDim.x + threadIdx.x;
  if (i >= n) return;
  int p = i ^ j;
  if (p <= i) return;
  unsigned long long a = d[i], b = d[p];
  bool up = ((i & k) == 0);                      // descending overall
  if (up ? (a < b) : (a > b)) { d[i] = b; d[p] = a; }
}

__global__ void __launch_bounds__(kTile)
bitonic_lds_kernel(unsigned long long* __restrict__ d, int k, int jstart) {
  __shared__ unsigned long long sk[kTile];
  int t  = threadIdx.x;
  int gi = blockIdx.x * kTile + t;

  async_ldst_b64(&sk[t], &d[gi]);   // async global->LDS, no VGPR round trip
  wait_async0();                    // this wave's ASYNCcnt == 0
  __syncthreads();                  // all waves' tiles resident

  for (int j = jstart; j > 0; j >>= 1) {
    int p = t ^ j;
    if (p > t) {
      unsigned long long a = sk[t], b = sk[p];
      bool up = ((gi & k) == 0);
      if (up ? (a < b) : (a > b)) { sk[t] = b; sk[p] = a; }
    }
    __syncthreads();
  }
  d[gi] = sk[t];
}

// ---------------- 3) decode top-2000 per level ------------------------------
__global__ void decode_kernel(const float* __restrict__ bbox,
                              const unsigned long long* __restrict__ keys,
                              float* __restrict__ cboxes,
                              float* __restrict__ cscores,
                              int lvlBase, int H, int W, float stride) {
  int s = blockIdx.x * blockDim.x + threadIdx.x;
  if (s >= kPre) return;
  int o = lvlBase + s;
  unsigned long long key = keys[s];
  if (key == 0ull) {                              // padded slot (level 4 tail)
    cscores[o] = -__builtin_inff();
    cboxes[o * 4 + 0] = 0.f; cboxes[o * 4 + 1] = 0.f;
    cboxes[o * 4 + 2] = 0.f; cboxes[o * 4 + 3] = 0.f;
    return;
  }
  float logit = key2f((unsigned)(key >> 32));
  unsigned tid = 0xFFFFFFFFu - (unsigned)(key & 0xFFFFFFFFull);
  int a = (int)(tid % kA);
  int r = (int)(tid / kA);
  int x = r % W;
  int y = r / W;

  int   hw = H * W;
  int   px_off = y * W + x;
  float dx = bbox[(4 * a + 0) * hw + px_off];
  float dy = bbox[(4 * a + 1) * hw + px_off];
  float dw = bbox[(4 * a + 2) * hw + px_off];
  float dh = bbox[(4 * a + 3) * hw + px_off];
  const float maxr = 4.135166556742356f;          // log(1000/16)
  dw = fminf(fmaxf(dw, -maxr), maxr);
  dh = fminf(fmaxf(dh, -maxr), maxr);

  // anchors on the fly: ratios {0.5,1,2}, scale 8, base_size = stride
  const float hr_tab[3] = {0.70710678118654752f, 1.f, 1.41421356237309515f};
  float hr = hr_tab[a];
  float wr = 1.f / hr;
  float pw = stride * wr * 8.f;                   // x2-x1+1 of base anchor
  float ph = stride * hr * 8.f;
  float ctr = 0.5f * (stride - 1.f);
  float px = (float)x * stride + ctr;
  float py = (float)y * stride + ctr;

  float gw = pw * expf(dw);
  float gh = ph * expf(dh);
  float gx = px + pw * dx;
  float gy = py + ph * dy;
  float x1 = fminf(fmaxf(gx - 0.5f * (gw - 1.f), 0.f), 1343.f);
  float y1 = fminf(fmaxf(gy - 0.5f * (gh - 1.f), 0.f), 799.f);
  float x2 = fminf(fmaxf(gx + 0.5f * (gw - 1.f), 0.f), 1343.f);
  float y2 = fminf(fmaxf(gy + 0.5f * (gh - 1.f), 0.f), 799.f);

  cboxes[o * 4 + 0] = x1; cboxes[o * 4 + 1] = y1;
  cboxes[o * 4 + 2] = x2; cboxes[o * 4 + 3] = y2;
  cscores[o] = 1.f / (1.f + expf(-logit));
}

// ---------------- 4) sort all candidates by score ---------------------------
__global__ void nms_keys_kernel(const float* __restrict__ cscores,
                                unsigned long long* __restrict__ nk) {
  int t = blockIdx.x * blockDim.x + threadIdx.x;
  if (t >= kNmsPad) return;
  unsigned long long key = 0ull;
  if (t < kNC)
    key = ((unsigned long long)f2key(cscores[t]) << 32) |
          (unsigned)(0xFFFFFFFFu - (unsigned)t);   // stable: lower idx first
  nk[t] = key;
}

__global__ void gather_sorted_kernel(const unsigned long long* __restrict__ nk,
                                     const float* __restrict__ cboxes,
                                     const float* __restrict__ cscores,
                                     float* __restrict__ sboxes,
                                     float* __restrict__ sscores) {
  int t = blockIdx.x * blockDim.x + threadIdx.x;
  if (t >= kNC) return;
  unsigned src = 0xFFFFFFFFu - (unsigned)(nk[t] & 0xFFFFFFFFull);
  sboxes[t * 4 + 0] = cboxes[src * 4 + 0];
  sboxes[t * 4 + 1] = cboxes[src * 4 + 1];
  sboxes[t * 4 + 2] = cboxes[src * 4 + 2];
  sboxes[t * 4 + 3] = cboxes[src * 4 + 3];
  sscores[t] = cscores[src];
}

// ---------------- 5a) validity bitmask (keep init = isfinite) ---------------
__global__ void valid_mask_kernel(const float* __restrict__ ss,
                                  unsigned* __restrict__ validMask) {
  int w = blockIdx.x * blockDim.x + threadIdx.x;
  if (w >= kKeepW) return;
  unsigned m = 0u;
  for (int b = 0; b < 32; ++b) {
    int j = w * 32 + b;
    if (j < kNC && ss[j] > -__builtin_inff()) m |= (1u << b);
  }
  validMask[w] = m;
}

// ---------------- 5b) suppression-mask build (whole-chip parallel) ----------
// One wave per (rowBlk, colBlk) tile, upper triangle only. Column boxes are
// staged into LDS via one async b128 per lane; each lane emits one mask word.
__global__ void __launch_bounds__(32)
nms_mask_kernel(const float* __restrict__ sb, unsigned* __restrict__ mask) {
  int rb = blockIdx.x;
  int cb = blockIdx.y;
  if (cb < rb) return;                      // triangular: j > i only

  __shared__ __align__(16) float cbox[32 * 4];
  int lane = threadIdx.x;
  int j0 = cb * 32;
  // stage 32 column boxes (512B) into LDS; OOB lanes read scratch tail (safe)
  async_ldst_b128(&cbox[lane * 4], &sb[(size_t)(j0 + lane) * 4]);
  wait_async0();
  __builtin_amdgcn_s_barrier();

  int i = rb * 32 + lane;
  if (i >= kNC) return;

  float ix1 = sb[i * 4 + 0], iy1 = sb[i * 4 + 1];
  float ix2 = sb[i * 4 + 2], iy2 = sb[i * 4 + 3];
  float iar = (ix2 - ix1 + 1.f) * (iy2 - iy1 + 1.f);

  unsigned m = 0u;
  #pragma unroll 4
  for (int c = 0; c < 32; ++c) {
    int j = j0 + c;
    if (j > i && j < kNC) {
      float jx1 = cbox[c * 4 + 0], jy1 = cbox[c * 4 + 1];
      float jx2 = cbox[c * 4 + 2], jy2 = cbox[c * 4 + 3];
      float xx1 = fmaxf(ix1, jx1), yy1 = fmaxf(iy1, jy1);
      float xx2 = fminf(ix2, jx2), yy2 = fminf(iy2, jy2);
      float inter = fmaxf(xx2 - xx1 + 1.f, 0.f) * fmaxf(yy2 - yy1 + 1.f, 0.f);
      float jar = (jx2 - jx1 + 1.f) * (jy2 - jy1 + 1.f);
      float iou = inter / (iar + jar - inter);
      if (iou > kIouThr) m |= (1u << c);
    }
  }
  mask[(size_t)i * kKeepW + cb] = m;
}

// ---------------- 5c) single-wave pipelined bitmask reduce ------------------
__device__ __forceinline__ void load_row(const unsigned* __restrict__ mask,
                                         int i, int lane,
                                         unsigned* buf, unsigned* diag) {
  int bb = i >> 5;
  #pragma unroll
  for (int s = 0; s < kSlots; ++s) {
    int w = s * 32 + lane;
    buf[s] = (w >= bb && w < kKeepW) ? mask[(size_t)i * kKeepW + w] : 0u;
  }
  *diag = mask[(size_t)i * kKeepW + bb];     // uniform (broadcast) load
}

__global__ void __launch_bounds__(32)
nms_reduce_kernel(const unsigned* __restrict__ mask,
                  const unsigned* __restrict__ validMask,
                  unsigned* __restrict__ keepOut) {
  __shared__ unsigned rem[kKeepW];
  int lane = threadIdx.x;
  for (int w = lane; w < kKeepW; w += 32) rem[w] = ~validMask[w];

  unsigned buf[kSlots], nbuf[kSlots];
  unsigned diag = 0u, ndiag = 0u;
  #pragma unroll
  for (int s = 0; s < kSlots; ++s) { buf[s] = 0u; nbuf[s] = 0u; }

  load_row(mask, 0, lane, buf, &diag);
  unsigned cur = rem[0];                      // uniform across the wave

  for (int i = 0; i < kNC; ++i) {
    int b = i >> 5, s = i & 31;
    // prefetch row i+1 unless it is already removed (monotone -> safe skip)
    if (i + 1 < kNC) {
      int nb = (i + 1) >> 5, ns = (i + 1) & 31;
      unsigned wv = (nb == b) ? cur : rem[nb];
      if (!((wv >> ns) & 1u)) load_row(mask, i + 1, lane, nbuf, &ndiag);
    }
    bool kept = !((cur >> s) & 1u);           // uniform branch
    if (kept) {
      #pragma unroll
      for (int t = 0; t < kSlots; ++t) {
        int w = t * 32 + lane;
        if (w < kKeepW) rem[w] |= buf[t];     // each lane owns its words
      }
      cur |= diag;                            // keep register copy coherent
    }
    #pragma unroll
    for (int t = 0; t < kSlots; ++t) buf[t] = nbuf[t];
    diag = ndiag;
    if (s == 31 && i + 1 < kNC) cur = rem[(i + 1) >> 5];  // next block word
  }

  for (int w = lane; w < kKeepW; w += 32) keepOut[w] = ~rem[w];
}

// ---------------- 6) final top-1000 -----------------------------------------
__global__ void final_keys_kernel(const unsigned* __restrict__ keep,
                                  const float* __restrict__ sscores,
                                  unsigned long long* __restrict__ fk) {
  int t = blockIdx.x * blockDim.x + threadIdx.x;
  if (t >= kNmsPad) return;
  unsigned long long key = 0ull;
  if (t < kNC) {
    bool kept = (keep[t >> 5] >> (t & 31)) & 1u;
    float s = kept ? sscores[t] : -__builtin_inff();
    key = ((unsigned long long)f2key(s) << 32) |
          (unsigned)(0xFFFFFFFFu - (unsigned)t);
  }
  fk[t] = key;
}

__global__ void write_out_kernel(const unsigned long long* __restrict__ fk,
                                 const float* __restrict__ sboxes,
                                 float* __restrict__ out) {
  int t = blockIdx.x * blockDim.x + threadIdx.x;
  if (t >= kMax) return;
  unsigned long long key = fk[t];
  unsigned idx = 0xFFFFFFFFu - (unsigned)(key & 0xFFFFFFFFull);
  out[t * 5 + 0] = sboxes[idx * 4 + 0];
  out[t * 5 + 1] = sboxes[idx * 4 + 1];
  out[t * 5 + 2] = sboxes[idx * 4 + 2];
  out[t * 5 + 3] = sboxes[idx * 4 + 3];
  out[t * 5 + 4] = key2f((unsigned)(key >> 32));   // exact score (or -inf)
}

// ---------------- host-side sort schedule -----------------------------------
static void sort_u64(unsigned long long* d, int n, hipStream_t stream) {
  for (int k = 2; k <= n; k <<= 1) {
    int j = k >> 1;
    while (j >= kTile) {                         // partner crosses LDS tile
      bitonic_global_kernel<<<n / 256, 256, 0, stream>>>(d, n, j, k);
      j >>= 1;
    }
    bitonic_lds_kernel<<<n / kTile, kTile, 0, stream>>>(d, k, j);
  }
}

extern "C" void kernel_launch(void* const* d_in, const int* in_sizes, int n_in,
                              void* d_out, int out_size, void* d_ws, size_t ws_size,
                              hipStream_t stream) {
  (void)in_sizes; (void)n_in; (void)out_size; (void)ws_size;

  // d_in order (setup_inputs dict order): cls0,bbox0,cls1,bbox1,...
  const float* cls[kLevels];
  const float* bbx[kLevels];
  for (int i = 0; i < kLevels; ++i) {
    cls[i] = (const float*)d_in[2 * i + 0];
    bbx[i] = (const float*)d_in[2 * i + 1];
  }

  // workspace layout (all 8-byte aligned); total ~16.0 MB
  char* ws = (char*)d_ws;
  unsigned long long* lvlKeys = (unsigned long long*)ws;        // 349184 u64
  size_t off = (size_t)kKeysTotal * 8;
  float* cboxes  = (float*)(ws + off);  off += (size_t)kNC * 4 * 4;   // 160000
  float* cscores = (float*)(ws + off);  off += (size_t)kNC * 4;       // 40000
  unsigned long long* nmsKeys = (unsigned long long*)(ws + off);
  off += (size_t)kNmsPad * 8;                                         // 131072
  float* sboxes  = (float*)(ws + off);  off += (size_t)kNC * 4 * 4;
  float* sscores = (float*)(ws + off);  off += (size_t)kNC * 4;
  unsigned* keep = (unsigned*)(ws + off); off += 1280;                // 313 u32
  unsigned long long* finKeys = (unsigned long long*)(ws + off);
  off += (size_t)kNmsPad * 8;
  unsigned* validMask = (unsigned*)(ws + off); off += 1280;           // 313 u32
  unsigned* mask = (unsigned*)(ws + off);
  off += (size_t)kNC * kKeepW * 4;                                    // 12.52 MB

  // 1) keys + per-level descending sort (top-2000 ends up at the front)
  for (int i = 0; i < kLevels; ++i) {
    unsigned long long* keys = lvlKeys + kLOff[i];
    make_keys_kernel<<<kP[i] / 256, 256, 0, stream>>>(
        cls[i], keys, kV[i], kP[i], kHh[i], kWw[i]);
    sort_u64(keys, kP[i], stream);
  }

  // 2) decode top-2000 per level into the 10000-candidate arrays
  for (int i = 0; i < kLevels; ++i) {
    decode_kernel<<<(kPre + 255) / 256, 256, 0, stream>>>(
        bbx[i], lvlKeys + kLOff[i], cboxes, cscores,
        i * kPre, kHh[i], kWw[i], kStride[i]);
  }

  // 3) global descending order by score (stable in index)
  nms_keys_kernel<<<kNmsPad / 256, 256, 0, stream>>>(cscores, nmsKeys);
  sort_u64(nmsKeys, kNmsPad, stream);
  gather_sorted_kernel<<<(kNC + 255) / 256, 256, 0, stream>>>(
      nmsKeys, cboxes, cscores, sboxes, sscores);

  // 4) two-phase NMS: parallel mask build + single-wave pipelined reduce
  valid_mask_kernel<<<(kKeepW + 255) / 256, 256, 0, stream>>>(sscores, validMask);
  dim3 mgrid(kKeepW, kKeepW);
  nms_mask_kernel<<<mgrid, 32, 0, stream>>>(sboxes, mask);
  nms_reduce_kernel<<<1, 32, 0, stream>>>(mask, validMask, keep);

  // 5) final top-1000 and output
  final_keys_kernel<<<kNmsPad / 256, 256, 0, stream>>>(keep, sscores, finKeys);
  sort_u64(finKeys, kNmsPad, stream);
  write_out_kernel<<<(kMax + 255) / 256, 256, 0, stream>>>(
      finKeys, sboxes, (float*)d_out);
}